// feature_net_29317446762758
// MI455X (gfx1250) — compile-verified
//
#include <hip/hip_runtime.h>
#include <hip/hip_bf16.h>
#include <math.h>

// CDNA5 / gfx1250: wave32, WMMA 16x16x4 f32.
typedef __attribute__((ext_vector_type(2))) float v2f;
typedef __attribute__((ext_vector_type(8))) float v8f;

#define LN_EPS 1e-5f
#define NEG_SLOPE 0.2f

// Broadcast lane 0's value to all lanes (EXEC is full here): v_readfirstlane_b32.
__device__ __forceinline__ float bcast0(float v) {
    return __int_as_float(__builtin_amdgcn_readfirstlane(__float_as_int(v)));
}

// One wave handles one point per loop iteration:
//   A   = feat tile (16 h-rows x 4 channels), built in 2 VGPRs
//   B_t = W^T tile for output channels [16t, 16t+16), preloaded (16 tiles x 2 VGPRs)
//   D_t = A x B_t  (v_wmma_f32_16x16x4_f32), then LN over h + LeakyReLU + max over h
__global__ __launch_bounds__(256) void feature_net_kernel(
    const float* __restrict__ x,       // [n, 16]   (b=1, f=1 squeezed)
    const float* __restrict__ coords,  // [2, n, 16]; only plane 0 used
    const float* __restrict__ W,       // [256, 4] row-major
    const float* __restrict__ gamma,   // [16]
    const float* __restrict__ beta,    // [16]
    float* __restrict__ out,           // [n, 256]
    int n)
{
    const int lane  = threadIdx.x & 31;
    const int wave  = threadIdx.x >> 5;
    const int wpb   = blockDim.x >> 5;
    const int gwave = blockIdx.x * wpb + wave;
    const int nwav  = gridDim.x * wpb;

    const int m  = lane & 15;   // row (A) / output channel within tile (B,D)
    const int hi = lane >> 4;   // half-wave selector

    // ---- Preload B = W^T tiles (constant across points; L2-resident, 4 KB) ----
    // B 4x16 f32 layout: V0 lanes0-15 K=0, lanes16-31 K=2; V1 lanes0-15 K=1, lanes16-31 K=3.
    v2f Bt[16];
#pragma unroll
    for (int t = 0; t < 16; ++t) {
        const int o  = t * 16 + m;
        const int k0 = hi ? 2 : 0;
        Bt[t].x = W[o * 4 + k0];
        Bt[t].y = W[o * 4 + k0 + 1];
    }
    // gamma/beta indexed by h = v + 8*hi for D-register v
    float gv[8], bv[8];
#pragma unroll
    for (int v = 0; v < 8; ++v) {
        gv[v] = gamma[hi * 8 + v];
        bv[v] = beta[hi * 8 + v];
    }

    for (int i = gwave; i < n; i += nwav) {
        // ---- Build A tile: 16x4 f32 in 2 VGPRs ----
        // Lanes 0-15: V0=K0 (x_vd), V1=K1 (x_cd). Lanes 16-31: V0=K2 (x_cd), V1=K3 (x_v0).
        const float xm = x[i * 16 + m];
        const float cm = coords[i * 16 + m];
        const float x0 = bcast0(xm);   // v_readfirstlane (lane 0 holds m=0)
        const float c0 = bcast0(cm);
        const float xvd = xm - x0;
        const float xcd = cm - c0;
        v2f A;
        A.x = hi ? xcd : xvd;   // K=2 : K=0
        A.y = hi ? x0  : xcd;   // K=3 : K=1

        // Process tiles in pairs so every store is a fully coalesced 128B
        // all-lanes b32 store: lanes 0-15 carry tile 2tp, lanes 16-31 tile 2tp+1.
#pragma unroll
        for (int tp = 0; tp < 8; ++tp) {
            float r[2];
#pragma unroll
            for (int u = 0; u < 2; ++u) {
                const int t = tp * 2 + u;
                v8f acc = {};
                // D[h, o] = A[16x4] x B_t[4x16]  (+ 0)
                acc = __builtin_amdgcn_wmma_f32_16x16x4_f32(
                    /*neg_a=*/false, A, /*neg_b=*/false, Bt[t],
                    /*c_mod=*/(short)0, acc, /*reuse_a=*/false, /*reuse_b=*/false);

                // ---- LayerNorm over h (column of D): 8 regs in-lane + half-wave swap ----
                float s = 0.f, ss = 0.f;
#pragma unroll
                for (int v = 0; v < 8; ++v) {
                    const float d = acc[v];
                    s  += d;
                    ss += d * d;
                }
                s  += __shfl_xor(s, 16, 32);
                ss += __shfl_xor(ss, 16, 32);
                const float mu   = s * (1.f / 16.f);
                const float var  = ss * (1.f / 16.f) - mu * mu;
                const float rstd = rsqrtf(var + LN_EPS);

                // ---- affine + LeakyReLU + max over h ----
                float mx = -INFINITY;
#pragma unroll
                for (int v = 0; v < 8; ++v) {
                    float y = (acc[v] - mu) * rstd * gv[v] + bv[v];
                    y = (y >= 0.f) ? y : NEG_SLOPE * y;
                    mx = fmaxf(mx, y);
                }
                // after this, BOTH halves hold the final per-column result
                mx = fmaxf(mx, __shfl_xor(mx, 16, 32));
                r[u] = mx;
            }
            // lanes 0-15 write tile 2tp (cols m), lanes 16-31 write tile 2tp+1:
            // 32 consecutive floats -> one fully coalesced 128B store.
            out[i * 256 + tp * 32 + lane] = hi ? r[1] : r[0];
        }
    }
}

extern "C" void kernel_launch(void* const* d_in, const int* in_sizes, int n_in,
                              void* d_out, int out_size, void* d_ws, size_t ws_size,
                              hipStream_t stream) {
    const float* x      = (const float*)d_in[0];  // [1, n, 16, 1]
    const float* coords = (const float*)d_in[1];  // [2, n, 16]
    const float* W      = (const float*)d_in[2];  // [256, 4]
    const float* gamma  = (const float*)d_in[3];  // [16]
    const float* beta   = (const float*)d_in[4];  // [16]
    float* out          = (float*)d_out;          // [1, 1, n, 256]

    const int n = in_sizes[0] / 16;

    // 256 threads = 8 waves/block; grid-stride so each wave amortizes the
    // W^T / gamma / beta preload over several points. 1024 blocks = 8192 waves.
    int blocks = (n + 7) / 8;
    if (blocks > 1024) blocks = 1024;
    if (blocks < 1) blocks = 1;
    feature_net_kernel<<<blocks, 256, 0, stream>>>(x, coords, W, gamma, beta, out, n);
}